// NormalizeCLAHE_29910152249770
// MI455X (gfx1250) — compile-verified
//
#include <hip/hip_runtime.h>

typedef float v2f __attribute__((ext_vector_type(2)));
typedef float v8f __attribute__((ext_vector_type(8)));

#define NTOT    33554432            // 32*1*1024*1024
#define IMG_W   1024
#define NBIMG   32
#define GRIDT   8
#define TDIM    128
#define PIXELS  16384               // 128*128
#define BINS    256
#define CLIPV   2560u               // max(int(40*16384//256),1)
#define NBLK1   2048
#define PER_BLK 16384               // NTOT / NBLK1
#define NTILES  2048                // 32 * 8 * 8

// ws layout (floats):
//   [0]=a (scale), [1]=b (bias)
//   [PART_OFF .. PART_OFF+4*NBLK1): per-block partials (min,max,sum,sumsq)
//   [LUT_OFF  .. LUT_OFF+NTILES*BINS): per-tile LUTs
#define PART_OFF 16
#define LUT_OFF  (PART_OFF + 4 * NBLK1)

// ---------- Pass 1: per-block min/max/sum/sumsq ----------
__global__ void k_reduce1(const float* __restrict__ x, float* __restrict__ ws) {
    __shared__ float smin[256], smax[256], ssum[256], ssq[256];
    int tid = threadIdx.x;
    size_t base = (size_t)blockIdx.x * PER_BLK;
    float vmin = 3.4e38f, vmax = -3.4e38f, s = 0.f, sq = 0.f;
    for (int i = tid; i < PER_BLK; i += 256) {
        float v = x[base + i];
        vmin = fminf(vmin, v); vmax = fmaxf(vmax, v);
        s += v; sq = fmaf(v, v, sq);
    }
    smin[tid] = vmin; smax[tid] = vmax; ssum[tid] = s; ssq[tid] = sq;
    __syncthreads();
    for (int st = 128; st > 0; st >>= 1) {
        if (tid < st) {
            smin[tid] = fminf(smin[tid], smin[tid + st]);
            smax[tid] = fmaxf(smax[tid], smax[tid + st]);
            ssum[tid] += ssum[tid + st];
            ssq [tid] += ssq [tid + st];
        }
        __syncthreads();
    }
    if (tid == 0) {
        ws[PART_OFF + 0 * NBLK1 + blockIdx.x] = smin[0];
        ws[PART_OFF + 1 * NBLK1 + blockIdx.x] = smax[0];
        ws[PART_OFF + 2 * NBLK1 + blockIdx.x] = ssum[0];
        ws[PART_OFF + 3 * NBLK1 + blockIdx.x] = ssq [0];
    }
}

// ---------- Pass 2: finalize scalars; fold whole affine chain into a*x+b ----------
__global__ void k_reduce2(float* __restrict__ ws) {
    __shared__ float  smin[256], smax[256];
    __shared__ double ssum[256], ssq[256];
    int tid = threadIdx.x;
    float vmin = 3.4e38f, vmax = -3.4e38f; double s = 0.0, sq = 0.0;
    for (int i = tid; i < NBLK1; i += 256) {
        vmin = fminf(vmin, ws[PART_OFF + 0 * NBLK1 + i]);
        vmax = fmaxf(vmax, ws[PART_OFF + 1 * NBLK1 + i]);
        s  += (double)ws[PART_OFF + 2 * NBLK1 + i];
        sq += (double)ws[PART_OFF + 3 * NBLK1 + i];
    }
    smin[tid] = vmin; smax[tid] = vmax; ssum[tid] = s; ssq[tid] = sq;
    __syncthreads();
    for (int st = 128; st > 0; st >>= 1) {
        if (tid < st) {
            smin[tid] = fminf(smin[tid], smin[tid + st]);
            smax[tid] = fmaxf(smax[tid], smax[tid + st]);
            ssum[tid] += ssum[tid + st];
            ssq [tid] += ssq [tid + st];
        }
        __syncthreads();
    }
    if (tid == 0) {
        double xmin = (double)smin[0], xmax = (double)smax[0];
        double N = (double)NTOT;
        double inv0 = 1.0 / (xmax - xmin);
        double m1 = (ssum[0] / N - xmin) * inv0;                 // mean of x1
        double varx = (ssq[0] - ssum[0] * ssum[0] / N) / (N - 1.0);
        double s1 = sqrt(varx) * inv0;                           // std(x1), ddof=1
        // x2 = (x1 - m1)/s1 ; min2 = (0-m1)/s1, max2 = (1-m1)/s1
        double min2 = (0.0 - m1) / s1, max2 = (1.0 - m1) / s1;
        double invr = 1.0 / (max2 - min2);
        double a = inv0 / s1 * invr;
        double b = ((-xmin * inv0 - m1) / s1 - min2) * invr;
        ws[0] = (float)a;
        ws[1] = (float)b;
    }
}

// ---------- Pass 3: per-tile histogram -> clip -> redistribute -> WMMA cumsum -> LUT ----------
__global__ void k_hist_lut(const float* __restrict__ x, float* __restrict__ ws) {
    __shared__ unsigned hist[BINS];
    __shared__ float    fh[BINS];
    __shared__ float    T16[16];
    __shared__ unsigned sred[256];
    int tid  = threadIdx.x;
    int tile = blockIdx.x;                 // (img, ty, tx) = tile layout of reference
    int bimg = tile >> 6;
    int ty   = (tile >> 3) & 7;
    int tx   = tile & 7;
    float a = ws[0], b = ws[1];

    hist[tid] = 0u;
    __syncthreads();

    size_t base = (size_t)bimg * (IMG_W * (size_t)IMG_W)
                + (size_t)(ty * TDIM) * IMG_W + (size_t)(tx * TDIM);
    for (int i = tid; i < PIXELS; i += 256) {
        int r = i >> 7, c = i & 127;
        float v = fmaf(x[base + (size_t)r * IMG_W + c], a, b);
        int bin = (int)(v * 256.0f);
        bin = bin < 0 ? 0 : (bin > 255 ? 255 : bin);
        atomicAdd(&hist[bin], 1u);         // LDS ds_add_u32, exact & deterministic
    }
    __syncthreads();

    unsigned hv = hist[tid]; if (hv > CLIPV) hv = CLIPV;
    sred[tid] = hv; __syncthreads();
    for (int st = 128; st > 0; st >>= 1) {
        if (tid < st) sred[tid] += sred[tid + st];
        __syncthreads();
    }
    unsigned excess   = PIXELS - sred[0];
    unsigned redist   = excess >> 8;
    unsigned residual = excess & 255u;
    fh[tid] = (float)(hv + redist + ((unsigned)tid < residual ? 1u : 0u));
    __syncthreads();

    if (tid < 16) {                        // row totals of H[16][16]
        float t = 0.f;
        for (int j = 0; j < 16; ++j) t += fh[tid * 16 + j];
        T16[tid] = t;
    }
    __syncthreads();

    // cumsum[row*16+col] = (H x U)[row][col] + (Lstrict x R)[row][col]
    // wave 0 only: EXEC all-1s within the wave (WMMA requirement).
    if (tid < 32) {
        int lm = tid & 15;                 // A row / B,C,D column
        int ks = (tid >> 4) * 2;           // K sub-offset per ISA 16x4/4x16 striping
        v8f acc = {};
        #pragma unroll
        for (int s4 = 0; s4 < 4; ++s4) {   // P = H x U  (U[k][n] = k<=n)
            int k0 = s4 * 4;
            v2f av, bv;
            av.x = fh[lm * 16 + k0 + ks];
            av.y = fh[lm * 16 + k0 + ks + 1];
            bv.x = (k0 + ks     <= lm) ? 1.0f : 0.0f;
            bv.y = (k0 + ks + 1 <= lm) ? 1.0f : 0.0f;
            acc = __builtin_amdgcn_wmma_f32_16x16x4_f32(
                      false, av, false, bv, (short)0, acc, false, false);
        }
        #pragma unroll
        for (int s4 = 0; s4 < 4; ++s4) {   // acc += Lstrict x R  (L[m][k]=k<m, R[k][n]=T[k])
            int k0 = s4 * 4;
            v2f av, bv;
            av.x = (k0 + ks     < lm) ? 1.0f : 0.0f;
            av.y = (k0 + ks + 1 < lm) ? 1.0f : 0.0f;
            bv.x = T16[k0 + ks];
            bv.y = T16[k0 + ks + 1];
            acc = __builtin_amdgcn_wmma_f32_16x16x4_f32(
                      false, av, false, bv, (short)0, acc, false, false);
        }
        float* lut = ws + LUT_OFF + (size_t)tile * BINS;
        #pragma unroll
        for (int v = 0; v < 8; ++v) {      // C/D layout: VGPR v -> row v+8*(lane>=16), col lm
            int row = v + 8 * (tid >> 4);
            int bin = row * 16 + lm;
            float lv = floorf(acc[v] * (255.0f / 16384.0f));
            lv = lv < 0.f ? 0.f : (lv > 255.f ? 255.f : lv);
            lut[bin] = lv;
        }
    }
}

// ---------- Pass 4: bilinear LUT interpolation, float4 vectorized ----------
__global__ void k_apply(const float* __restrict__ x, const float* __restrict__ ws,
                        float* __restrict__ out) {
    size_t t = (size_t)blockIdx.x * blockDim.x + threadIdx.x;
    size_t p = t * 4;
    if (p >= (size_t)NTOT) return;
    float a = ws[0], b = ws[1];
    const float* lut = ws + LUT_OFF;

    float4 xv = *(const float4*)(x + p);
    float xi[4]; *(float4*)xi = xv;

    int bimg = (int)(p >> 20);             // H*W = 2^20
    int h    = (int)(p >> 10) & 1023;
    float fy = (h + 0.5f) * (1.0f / 128.0f) - 0.5f;
    fy = fy < 0.f ? 0.f : (fy > 7.f ? 7.f : fy);
    int gy0 = (int)fy; if (gy0 > 6) gy0 = 6;
    float wy = fy - (float)gy0;
    int rowbase = bimg * 64 + gy0 * 8;

    float res[4];
    #pragma unroll
    for (int i = 0; i < 4; ++i) {
        int w = ((int)p & 1023) + i;
        float fx = (w + 0.5f) * (1.0f / 128.0f) - 0.5f;
        fx = fx < 0.f ? 0.f : (fx > 7.f ? 7.f : fx);
        int gx0 = (int)fx; if (gx0 > 6) gx0 = 6;
        float wx = fx - (float)gx0;

        float v = fmaf(xi[i], a, b);
        int bin = (int)(v * 256.0f);
        bin = bin < 0 ? 0 : (bin > 255 ? 255 : bin);

        const float* l0 = lut + (size_t)(rowbase + gx0) * BINS;       // (gy0,gx0),(gy0,gx1)
        const float* l1 = lut + (size_t)(rowbase + 8 + gx0) * BINS;   // (gy1,gx0),(gy1,gx1)
        float v00 = l0[bin], v01 = l0[BINS + bin];
        float v10 = l1[bin], v11 = l1[BINS + bin];
        float top = v00 + wx * (v01 - v00);
        float bot = v10 + wx * (v11 - v10);
        res[i] = (top + wy * (bot - top)) * (1.0f / 255.0f);
    }
    *(float4*)(out + p) = *(float4*)res;
}

extern "C" void kernel_launch(void* const* d_in, const int* in_sizes, int n_in,
                              void* d_out, int out_size, void* d_ws, size_t ws_size,
                              hipStream_t stream) {
    (void)in_sizes; (void)n_in; (void)out_size; (void)ws_size;
    const float* x = (const float*)d_in[0];
    const int*   y = (const int*)d_in[1];
    float* ws  = (float*)d_ws;
    float* out = (float*)d_out;

    k_reduce1 <<<NBLK1,      256, 0, stream>>>(x, ws);
    k_reduce2 <<<1,          256, 0, stream>>>(ws);
    k_hist_lut<<<NTILES,     256, 0, stream>>>(x, ws);
    k_apply   <<<NTOT / 1024, 256, 0, stream>>>(x, ws, out);

    // y passes through unchanged (bitwise copy preserves int32 exactly)
    hipMemcpyAsync((char*)d_out + (size_t)NTOT * sizeof(float), y,
                   (size_t)NTOT * sizeof(int), hipMemcpyDeviceToDevice, stream);
}